// FWI_costfunction_17222818857434
// MI455X (gfx1250) — compile-verified
//
#include <hip/hip_runtime.h>
#include <hip/hip_bf16.h>
#include <cstdint>

// Problem geometry (fixed by the reference): [4, 16, 4000, 400] fp32.
#define PLANES      64          // batch * shot
#define PLANE_ELEMS 1600000     // 4000 * 400
#define PLANE_F4    400000      // float4 per plane
#define BPP         40          // blocks per plane
#define TPB         256         // threads per block (8 wave32 waves)
#define NWAVES      (TPB / 32)
#define BLOCK_F4    (PLANE_F4 / BPP)        // 10000 float4 per block
#define WAVE_F4     (BLOCK_F4 / NWAVES)     // 1250 float4 per wave
#define NSTAGES     ((WAVE_F4 + 31) / 32)   // 40 stages of 32 float4
#define DEPTH       4                       // async pipeline depth (stages)

// Low 32 bits of a generic pointer to a __shared__ object == LDS byte offset
// (flat LDS aperture keeps the offset in addr[31:0]; HW truncates).
__device__ __forceinline__ unsigned lds_off(const void* p) {
  return (unsigned)(uintptr_t)p;
}

// CDNA5 async copy: global -> LDS, 16 bytes per active lane, ASYNCcnt-tracked.
// NT temporal hint: stream-once data, don't retain in L2.
#define ASYNC_CP_B128(ldsoff, gptr)                                        \
  asm volatile("global_load_async_to_lds_b128 %0, %1, off th:TH_LOAD_NT"   \
               :                                                           \
               : "v"(ldsoff), "v"((unsigned long long)(uintptr_t)(gptr))   \
               : "memory")

#define WAIT_ASYNC(n) asm volatile("s_wait_asynccnt " #n ::: "memory")

__global__ __launch_bounds__(TPB) void fwi_corr_partial(
    const float* __restrict__ X,   // shot_syn
    const float* __restrict__ Y,   // shot_obs
    float* __restrict__ ws)        // [PLANES*BPP][5] block partials
{
  // Per-wave private 4-deep staging ring: no block barriers in the streaming
  // loop; each wave's own ASYNCcnt orders its own copies (in-order retire).
  __shared__ float4 bufX[NWAVES][DEPTH][32];   // 16 KB
  __shared__ float4 bufY[NWAVES][DEPTH][32];   // 16 KB
  __shared__ float  red[NWAVES][5];

  const int plane = blockIdx.x / BPP;
  const int blk   = blockIdx.x % BPP;
  const int wave  = threadIdx.x >> 5;
  const int lane  = threadIdx.x & 31;

  const float4* xp = (const float4*)(X + (long long)plane * PLANE_ELEMS);
  const float4* yp = (const float4*)(Y + (long long)plane * PLANE_ELEMS);

  const int wstart = blk * BLOCK_F4 + wave * WAVE_F4;  // float4 index in plane
  const int wend   = wstart + WAVE_F4;

  // 20 fp32 accumulators (5 sums x 4 vector lanes) — ILP + precision.
  float sx0=0,sx1=0,sx2=0,sx3=0;
  float sy0=0,sy1=0,sy2=0,sy3=0;
  float xy0=0,xy1=0,xy2=0,xy3=0;
  float xx0=0,xx1=0,xx2=0,xx3=0;
  float yy0=0,yy1=0,yy2=0,yy3=0;

  auto issue = [&](int s) {
    const int i = wstart + s * 32 + lane;
    if (i < wend) {  // ragged tail: lanes 0..1 of stage 39 (same on all waves)
      ASYNC_CP_B128(lds_off(&bufX[wave][s & (DEPTH - 1)][lane]), xp + i);
      ASYNC_CP_B128(lds_off(&bufY[wave][s & (DEPTH - 1)][lane]), yp + i);
    }
  };
  auto consume = [&](int s) {
    const int i = wstart + s * 32 + lane;
    if (i < wend) {
      const float4 xv = bufX[wave][s & (DEPTH - 1)][lane];  // ds_load_b128
      const float4 yv = bufY[wave][s & (DEPTH - 1)][lane];
      sx0 += xv.x; sx1 += xv.y; sx2 += xv.z; sx3 += xv.w;
      sy0 += yv.x; sy1 += yv.y; sy2 += yv.z; sy3 += yv.w;
      xy0 = fmaf(xv.x, yv.x, xy0); xy1 = fmaf(xv.y, yv.y, xy1);
      xy2 = fmaf(xv.z, yv.z, xy2); xy3 = fmaf(xv.w, yv.w, xy3);
      xx0 = fmaf(xv.x, xv.x, xx0); xx1 = fmaf(xv.y, xv.y, xx1);
      xx2 = fmaf(xv.z, xv.z, xx2); xx3 = fmaf(xv.w, xv.w, xx3);
      yy0 = fmaf(yv.x, yv.x, yy0); yy1 = fmaf(yv.y, yv.y, yy1);
      yy2 = fmaf(yv.z, yv.z, yy2); yy3 = fmaf(yv.w, yv.w, yy3);
    }
  };

  // Prologue: fill the pipeline (NSTAGES=40 >= DEPTH=4). 8 copies in flight.
  issue(0); issue(1); issue(2); issue(3);

  // Steady state: before the wait, stages s..s+3 are outstanding (8 async
  // ops). asynccnt<=6 guarantees stage s has fully landed in LDS (in-order
  // completion); consume it, then refill with stage s+4.
  for (int s = 0; s < NSTAGES - DEPTH; ++s) {
    WAIT_ASYNC(0x6);
    consume(s);
    issue(s + DEPTH);
  }
  // Drain: outstanding count shrinks 8 -> 6 -> 4 -> 2.
  WAIT_ASYNC(0x6); consume(NSTAGES - 4);
  WAIT_ASYNC(0x4); consume(NSTAGES - 3);
  WAIT_ASYNC(0x2); consume(NSTAGES - 2);
  WAIT_ASYNC(0x0); consume(NSTAGES - 1);

  float sx  = (sx0 + sx1) + (sx2 + sx3);
  float sy  = (sy0 + sy1) + (sy2 + sy3);
  float sxy = (xy0 + xy1) + (xy2 + xy3);
  float sxx = (xx0 + xx1) + (xx2 + xx3);
  float syy = (yy0 + yy1) + (yy2 + yy3);

  // wave32 butterfly reduction.
#pragma unroll
  for (int o = 16; o > 0; o >>= 1) {
    sx  += __shfl_xor(sx,  o, 32);
    sy  += __shfl_xor(sy,  o, 32);
    sxy += __shfl_xor(sxy, o, 32);
    sxx += __shfl_xor(sxx, o, 32);
    syy += __shfl_xor(syy, o, 32);
  }
  if (lane == 0) {
    red[wave][0] = sx;  red[wave][1] = sy;  red[wave][2] = sxy;
    red[wave][3] = sxx; red[wave][4] = syy;
  }
  __syncthreads();
  if (threadIdx.x == 0) {
    float a0 = 0, a1 = 0, a2 = 0, a3 = 0, a4 = 0;
#pragma unroll
    for (int w = 0; w < NWAVES; ++w) {
      a0 += red[w][0]; a1 += red[w][1]; a2 += red[w][2];
      a3 += red[w][3]; a4 += red[w][4];
    }
    float* o = ws + (size_t)blockIdx.x * 5;
    o[0] = a0; o[1] = a1; o[2] = a2; o[3] = a3; o[4] = a4;
  }
}

__global__ __launch_bounds__(64) void fwi_corr_finalize(
    const float* __restrict__ ws, float* __restrict__ out)
{
  __shared__ double wsum[2];
  const int p = threadIdx.x;  // one thread per plane (0..63)

  double sx = 0, sy = 0, sxy = 0, sxx = 0, syy = 0;
  const float* q = ws + (size_t)p * BPP * 5;
#pragma unroll 4
  for (int b = 0; b < BPP; ++b) {   // fixed order -> deterministic
    sx  += (double)q[0];
    sy  += (double)q[1];
    sxy += (double)q[2];
    sxx += (double)q[3];
    syy += (double)q[4];
    q += 5;
  }
  const double N = (double)PLANE_ELEMS;
  const double cross = sxy - sx * sy / N;   // sum((x-mx)(y-my))
  const double ep    = sxx - sx * sx / N;   // sum((x-mx)^2)
  const double et    = syy - sy * sy / N;   // sum((y-my)^2)
  double r = cross / (sqrt(et) * sqrt(ep));

#pragma unroll
  for (int o = 16; o > 0; o >>= 1) r += __shfl_xor(r, o, 32);
  if ((p & 31) == 0) wsum[p >> 5] = r;
  __syncthreads();
  if (p == 0) out[0] = (float)(-(wsum[0] + wsum[1]));
}

extern "C" void kernel_launch(void* const* d_in, const int* in_sizes, int n_in,
                              void* d_out, int out_size, void* d_ws, size_t ws_size,
                              hipStream_t stream) {
  const float* X = (const float*)d_in[0];   // shot_syn
  const float* Y = (const float*)d_in[1];   // shot_obs
  float* ws  = (float*)d_ws;                // 64*40*5 floats = 51.2 KB
  float* out = (float*)d_out;

  fwi_corr_partial<<<PLANES * BPP, TPB, 0, stream>>>(X, Y, ws);
  fwi_corr_finalize<<<1, 64, 0, stream>>>(ws, out);
}